// ConvLayer_37778532335652
// MI455X (gfx1250) — compile-verified
//
#include <hip/hip_runtime.h>

typedef __attribute__((ext_vector_type(2))) float v2f;
typedef __attribute__((ext_vector_type(8))) float v8f;

#define IN_DIM 64
#define OUT_DIM 64

// ---------------------------------------------------------------------------
// Kernel 1: zero the aggregation workspace (float4 stores, one pass)
// ---------------------------------------------------------------------------
__global__ void gcn_zero_ws(float4* __restrict__ agg4, int n4) {
    int i = blockIdx.x * blockDim.x + threadIdx.x;
    if (i < n4) agg4[i] = make_float4(0.f, 0.f, 0.f, 0.f);
}

// ---------------------------------------------------------------------------
// Kernel 2: SpMM scatter. 16 threads per edge, each handles 4 dims (float4
// gather of x + 4 HW f32 atomic adds into agg). Atomics land in L2; x (25.6MB)
// is L2-resident, so gathers mostly hit.
// ---------------------------------------------------------------------------
__global__ void gcn_spmm_scatter(const float* __restrict__ x,
                                 const float* __restrict__ vals,
                                 const long long* __restrict__ row,
                                 const long long* __restrict__ col,
                                 float* __restrict__ agg,
                                 int nEdges) {
    int gid = blockIdx.x * blockDim.x + threadIdx.x;
    int e   = gid >> 4;          // edge index
    int c4  = (gid & 15) << 2;   // dim offset: 0,4,...,60
    if (e >= nEdges) return;

    long long r = row[e];
    long long c = col[e];
    float     v = vals[e];

    const float4 xv = *(const float4*)(x + c * (long long)IN_DIM + c4);
    float* dst = agg + r * (long long)IN_DIM + c4;

    unsafeAtomicAdd(dst + 0, v * xv.x);   // global_atomic_add_f32
    unsafeAtomicAdd(dst + 1, v * xv.y);
    unsafeAtomicAdd(dst + 2, v * xv.z);
    unsafeAtomicAdd(dst + 3, v * xv.w);
}

// ---------------------------------------------------------------------------
// Kernel 3: out = agg @ W + bias via V_WMMA_F32_16X16X4_F32.
// One wave per 16-row tile of agg; N=64 handled as 4 column tiles of 16.
// Bias is folded in as the initial accumulator value.
// Block = 320 threads = 10 waves; grid = 625 blocks -> exactly 6250 tiles.
// ---------------------------------------------------------------------------
__global__ void __launch_bounds__(320)
gcn_gemm_wmma(const float* __restrict__ agg,
              const float* __restrict__ W,
              const float* __restrict__ bias,
              float* __restrict__ out,
              int nRowTiles) {
    const int lane  = threadIdx.x & 31;
    const int wave  = threadIdx.x >> 5;
    const int tile  = blockIdx.x * 10 + wave;
    if (tile >= nRowTiles) return;           // wave-uniform: EXEC stays all-1s

    const int rowbase = tile * 16;
    const int mlane   = lane & 15;           // row within tile (A/M, or N col)
    const int khalf   = (lane >> 4) * 2;     // lanes 16-31 hold K+2 / K+3

    // --- Preload all 16 A fragments (16x4 f32 each): a.x/a.y contiguous ---
    const float* arow = agg + (long long)(rowbase + mlane) * IN_DIM;
    v2f a[16];
#pragma unroll
    for (int kk = 0; kk < 16; ++kk)
        a[kk] = *(const v2f*)(arow + kk * 4 + khalf);

    // --- 4 column tiles of 16 ---
#pragma unroll
    for (int nt = 0; nt < 4; ++nt) {
        const int nb = nt * 16 + mlane;      // global N column for this lane
        const float bv = bias[nb];
        v8f c = { bv, bv, bv, bv, bv, bv, bv, bv };  // fold bias into C

#pragma unroll
        for (int kk = 0; kk < 16; ++kk) {
            const int k = kk * 4 + khalf;
            v2f bfrag;
            bfrag.x = W[(k + 0) * OUT_DIM + nb];   // B row K,   col nb
            bfrag.y = W[(k + 1) * OUT_DIM + nb];   // B row K+1, col nb
            c = __builtin_amdgcn_wmma_f32_16x16x4_f32(
                    /*neg_a=*/false, a[kk],
                    /*neg_b=*/false, bfrag,
                    /*c_mod=*/(short)0, c,
                    /*reuse_a=*/false, /*reuse_b=*/false);
        }

        // --- Store D: VGPR v -> row (rowbase + v + 8*(lane>=16)), col nb ---
        float* op = out + (long long)(rowbase + (lane >> 4) * 8) * OUT_DIM + nb;
#pragma unroll
        for (int v = 0; v < 8; ++v)
            op[v * OUT_DIM] = c[v];
    }
}

// ---------------------------------------------------------------------------
// Inputs (setup_inputs order): x[N,64] f32, adj_vals[E] f32, adj_row[E] i64,
// adj_col[E] i64, W[64,64] f32, b[64] f32.  Output: [N,64] f32.
// d_ws: agg buffer, N*64*4 = 25.6 MB.
// ---------------------------------------------------------------------------
extern "C" void kernel_launch(void* const* d_in, const int* in_sizes, int n_in,
                              void* d_out, int out_size, void* d_ws, size_t ws_size,
                              hipStream_t stream) {
    const float*     x    = (const float*)d_in[0];
    const float*     vals = (const float*)d_in[1];
    const long long* row  = (const long long*)d_in[2];
    const long long* col  = (const long long*)d_in[3];
    const float*     W    = (const float*)d_in[4];
    const float*     bias = (const float*)d_in[5];
    float*           out  = (float*)d_out;
    float*           agg  = (float*)d_ws;

    const int nNodes = in_sizes[0] / IN_DIM;   // 100000
    const int nEdges = in_sizes[1];            // 1000000

    // 1) zero agg
    const int n4 = nNodes * IN_DIM / 4;
    gcn_zero_ws<<<(n4 + 255) / 256, 256, 0, stream>>>((float4*)agg, n4);

    // 2) scatter: 16 threads per edge
    const long long nThreads = (long long)nEdges * 16;
    gcn_spmm_scatter<<<(unsigned)((nThreads + 255) / 256), 256, 0, stream>>>(
        x, vals, row, col, agg, nEdges);

    // 3) WMMA GEMM + bias. nNodes = 100000 is divisible by 16 -> 6250 tiles.
    const int nRowTiles = nNodes / 16;
    gcn_gemm_wmma<<<(nRowTiles + 9) / 10, 320, 0, stream>>>(agg, W, bias, out,
                                                            nRowTiles);
}